// RollingAvg_58815282151635
// MI455X (gfx1250) — compile-verified
//
#include <hip/hip_runtime.h>

// Rolling-average forecast, linearized:  Out[32 x N] = C[32 x 10] * W[10 x N]
// N = 4096*128 = 524288 positions, W = x[54:64] (last RA_WINDOW slices).
// Memory-bound (88 MB traffic, ~3.8us at 23.3 TB/s); f32 WMMA used since
// matrix throughput is free at this arithmetic intensity and keeps f32 accuracy.

#define RA_WINDOW 10
#define HORIZON   32
#define KPAD      12                    // 10 padded to 3 WMMA K=4 steps
#define NPOS      (4096LL * 128LL)      // positions per horizon step

typedef __attribute__((ext_vector_type(2))) float v2f;
typedef __attribute__((ext_vector_type(8))) float v8f;

struct Coefs { float c[HORIZON][KPAD]; };

// Compile-time simulation of the scan over basis vectors -> coefficient matrix.
constexpr Coefs make_coefs() {
    Coefs out{};
    double win[RA_WINDOW][RA_WINDOW] = {};      // win[slot][k]: coeff of w_k
    for (int i = 0; i < RA_WINDOW; ++i) win[i][i] = 1.0;
    for (int h = 0; h < HORIZON; ++h) {
        double nc[RA_WINDOW] = {};
        for (int k = 0; k < RA_WINDOW; ++k) {
            double s = 0.0;
            for (int i = 0; i < RA_WINDOW; ++i) s += win[i][k];
            nc[k] = s / (double)RA_WINDOW;
        }
        for (int k = 0; k < KPAD; ++k)
            out.c[h][k] = (k < RA_WINDOW) ? (float)nc[k] : 0.0f;
        for (int i = 0; i < RA_WINDOW - 1; ++i)
            for (int k = 0; k < RA_WINDOW; ++k) win[i][k] = win[i + 1][k];
        for (int k = 0; k < RA_WINDOW; ++k) win[RA_WINDOW - 1][k] = nc[k];
    }
    return out;
}

constexpr Coefs HOST_COEFS = make_coefs();
__constant__ Coefs COEFS = HOST_COEFS;

__global__ __launch_bounds__(256)
void RollingAvg_wmma_kernel(const float* __restrict__ x, float* __restrict__ out) {
    const int lane = threadIdx.x & 31;
    const int col  = lane & 15;     // N within tile
    const int half = lane >> 4;     // half-wave select

    // One 16-wide N-tile per wave; grid sized exactly: NPOS/16 waves.
    const long long wave = (long long)blockIdx.x * (blockDim.x >> 5) + (threadIdx.x >> 5);
    const long long n0   = wave * 16;

    const float* W = x + 54LL * NPOS;   // x[54..63] = last RA_WINDOW slices

    // ---- B fragments (W tile, K x 16 per K-step) ----
    // ISA 32-bit split-half layout assumption: VGPR j, lanes 0-15 -> k = 4s+j,
    // lanes 16-31 -> k = 4s+j+2. k >= 10 is zero padding (beyond x!).
    v2f b[3];
#pragma unroll
    for (int s = 0; s < 3; ++s) {
#pragma unroll
        for (int j = 0; j < 2; ++j) {
            const int k = 4 * s + j + 2 * half;
            b[s][j] = (k < RA_WINDOW) ? W[(long long)k * NPOS + n0 + col] : 0.0f;
        }
    }

    // ---- A fragments from compile-time coefficient table ----
    // ISA 16x4 f32 A layout: lane m = lane&15 (both halves);
    // VGPR j: k = 4s + j (lanes 0-15), k = 4s + j + 2 (lanes 16-31).
    v8f d0 = {0.f,0.f,0.f,0.f,0.f,0.f,0.f,0.f};   // horizons 0..15
    v8f d1 = {0.f,0.f,0.f,0.f,0.f,0.f,0.f,0.f};   // horizons 16..31
#pragma unroll
    for (int s = 0; s < 3; ++s) {
        v2f a0, a1;
#pragma unroll
        for (int j = 0; j < 2; ++j) {
            const int k = 4 * s + j + 2 * half;
            a0[j] = COEFS.c[col][k];
            a1[j] = COEFS.c[16 + col][k];
        }
        // 8 args: (neg_a, A, neg_b, B, c_mod, C, reuse_a, reuse_b)
        d0 = __builtin_amdgcn_wmma_f32_16x16x4_f32(false, a0, false, b[s],
                                                   (short)0, d0, false, false);
        d1 = __builtin_amdgcn_wmma_f32_16x16x4_f32(false, a1, false, b[s],
                                                   (short)0, d1, false, false);
    }

    // ---- Store D: VGPR r, lanes 0-15 -> (m=r, n=col), lanes 16-31 -> (m=r+8, n=col).
    // preds[h, pos] flat = h*NPOS + pos.
#pragma unroll
    for (int r = 0; r < 8; ++r) {
        const int m = r + 8 * half;
        out[(long long)m        * NPOS + n0 + col] = d0[r];
        out[(long long)(16 + m) * NPOS + n0 + col] = d1[r];
    }
}

extern "C" void kernel_launch(void* const* d_in, const int* in_sizes, int n_in,
                              void* d_out, int out_size, void* d_ws, size_t ws_size,
                              hipStream_t stream) {
    (void)in_sizes; (void)n_in; (void)d_ws; (void)ws_size; (void)out_size;
    const float* x = (const float*)d_in[0];
    float* out = (float*)d_out;

    // NPOS/16 = 32768 waves -> 256 threads (8 waves) per block -> 4096 blocks.
    const int threads = 256;
    const int wavesPerBlock = threads / 32;
    const int totalWaves = (int)(NPOS / 16);
    const int blocks = totalWaves / wavesPerBlock;   // 4096

    RollingAvg_wmma_kernel<<<blocks, threads, 0, stream>>>(x, out);
}